// Net_for_classification2_61357902791129
// MI455X (gfx1250) — compile-verified
//
#include <hip/hip_runtime.h>

#define N_NODES 100000
#define N_EDGES 3200000
#define D_FEAT  128
#define N1      128
#define N2      64
#define NCLS    10
#define NGRAPH  256

typedef __attribute__((ext_vector_type(2))) float v2f;
typedef __attribute__((ext_vector_type(4))) float v4f;
typedef __attribute__((ext_vector_type(8))) float v8f;

// ---------- init ----------
__global__ void k_fill4(float* __restrict__ p, long n4, float v) {
  // n4 = count of float4 elements; emits global_store_b128
  long i = (long)blockIdx.x * blockDim.x + threadIdx.x;
  long stride = (long)gridDim.x * blockDim.x;
  v4f val = {v, v, v, v};
  for (; i < n4; i += stride) ((v4f*)p)[i] = val;
}

// ---------- degree / normalization ----------
__global__ void k_degree(const int* __restrict__ ei, float* __restrict__ deg) {
  int i = blockIdx.x * blockDim.x + threadIdx.x;
  int stride = gridDim.x * blockDim.x;
  for (; i < N_EDGES; i += stride)
    atomicAdd(&deg[ei[N_EDGES + i]], 1.0f);
}

__global__ void k_invsqrt(float* __restrict__ deg) {
  int i = blockIdx.x * blockDim.x + threadIdx.x;
  if (i < N_NODES) deg[i] = rsqrtf(deg[i]);  // deg >= 1 always (self-loop)
}

// ---------- GEMM: C[M,N] = A[M,K] @ B[K,N], row-major f32 ----------
// One wave per 16x16 output tile using V_WMMA_F32_16X16X4_F32.
__global__ void k_gemm_wmma(const float* __restrict__ A, const float* __restrict__ B,
                            float* __restrict__ C, int M, int K, int N) {
  int wave = (int)((blockIdx.x * (unsigned)blockDim.x + threadIdx.x) >> 5);
  int lane = threadIdx.x & 31;
  int tilesN = N >> 4;
  int tm = wave / tilesN;
  int tn = wave - tm * tilesN;
  if (tm * 16 >= M) return;

  int r  = lane & 15;          // row (A) / col (B) within tile
  int kh = (lane >> 4) << 1;   // lanes 0-15: K={k,k+1}; lanes 16-31: K={k+2,k+3}
  const float* arow = A + (size_t)(tm * 16 + r) * K;
  const float* bcol = B + tn * 16 + r;

  v8f acc = {};
  for (int k = 0; k < K; k += 4) {
    v2f a, b;
    a.x = arow[k + kh];
    a.y = arow[k + kh + 1];
    b.x = bcol[(size_t)(k + kh) * N];
    b.y = bcol[(size_t)(k + kh + 1) * N];
    acc = __builtin_amdgcn_wmma_f32_16x16x4_f32(false, a, false, b,
                                                (short)0, acc, false, false);
  }

  // C/D layout: VGPR i -> row i (lanes 0-15) or row i+8 (lanes 16-31)
  float* crow = C + (size_t)(tm * 16 + ((lane >> 4) << 3)) * N + tn * 16 + r;
#pragma unroll
  for (int i = 0; i < 8; ++i) crow[(size_t)i * N] = acc[i];
}

// ---------- edge scatter: agg[dst] += h[src] * dis[src]*dis[dst] ----------
// One wave per edge. Each lane gathers VEC contiguous floats with one
// global_load_b128/b64 and scatters them with f32 atomics (L2-resident).
template <int D>
__global__ void k_scatter(const int* __restrict__ ei, const float* __restrict__ dis,
                          const float* __restrict__ h, float* __restrict__ agg) {
  constexpr int VEC = D / 32;  // floats per lane (4 for D=128, 2 for D=64)
  int lane = threadIdx.x & 31;
  long w  = ((long)blockIdx.x * blockDim.x + threadIdx.x) >> 5;
  long nw = ((long)gridDim.x * blockDim.x) >> 5;
  for (long e = w; e < N_EDGES; e += nw) {
    int s = ei[e];
    int d = ei[N_EDGES + e];
    // prefetch next edge's source row while we chew on this one
    long en = e + nw;
    if (en < N_EDGES) {
      int sn = ei[en];
      __builtin_prefetch(h + (size_t)sn * D + lane * VEC, 0, 1);
    }
    float norm = dis[s] * dis[d];
    const float* hs = h + (size_t)s * D + lane * VEC;
    float* ad = agg + (size_t)d * D + lane * VEC;
    float vals[VEC];
#pragma unroll
    for (int j = 0; j < VEC; ++j) vals[j] = hs[j];  // merges to b128/b64 load
#pragma unroll
    for (int j = 0; j < VEC; ++j) atomicAdd(&ad[j], vals[j] * norm);
  }
}

// ---------- agg[i,d] += h[i,d]*dis[i]^2 + bias[d]; optional ReLU ----------
template <int D, bool RELU>
__global__ void k_finalize(const float* __restrict__ h, const float* __restrict__ dis,
                           const float* __restrict__ bias, float* __restrict__ agg) {
  long idx = (long)blockIdx.x * blockDim.x + threadIdx.x;  // in float4 units
  long n4 = (long)N_NODES * (D / 4);
  long stride = (long)gridDim.x * blockDim.x;
  for (; idx < n4; idx += stride) {
    int node = (int)(idx / (D / 4));
    int d4   = (int)(idx - (long)node * (D / 4));
    float di = dis[node];
    float ds2 = di * di;
    v4f hv = ((const v4f*)h)[idx];
    v4f av = ((v4f*)agg)[idx];
    v4f bv = ((const v4f*)bias)[d4];
#pragma unroll
    for (int j = 0; j < 4; ++j) {
      float v = av[j] + hv[j] * ds2 + bv[j];
      av[j] = RELU ? fmaxf(v, 0.0f) : v;
    }
    ((v4f*)agg)[idx] = av;
  }
}

// ---------- pooling + FC ----------
__global__ void k_pool_cnt(const int* __restrict__ batch, float* __restrict__ pcnt) {
  int i = blockIdx.x * blockDim.x + threadIdx.x;
  if (i < N_NODES) atomicAdd(&pcnt[batch[i]], 1.0f);
}

__global__ void k_pool_sum(const int* __restrict__ batch, const float* __restrict__ feat,
                           float* __restrict__ psum) {
  long idx = (long)blockIdx.x * blockDim.x + threadIdx.x;
  long n = (long)N_NODES * N2;
  if (idx < n) {
    int node = (int)(idx >> 6);   // / N2 (=64)
    int d    = (int)(idx & 63);
    atomicAdd(&psum[batch[node] * N2 + d], feat[idx]);
  }
}

__global__ void k_fc(const float* __restrict__ psum, const float* __restrict__ pcnt,
                     const float* __restrict__ Wfc, const float* __restrict__ bfc,
                     float* __restrict__ out) {
  int i = blockIdx.x * blockDim.x + threadIdx.x;  // g*NCLS + c
  if (i < NGRAPH * NCLS) {
    int g = i / NCLS, c = i - g * NCLS;
    float inv = 1.0f / fmaxf(pcnt[g], 1.0f);
    float acc = bfc[c];
#pragma unroll 8
    for (int d = 0; d < N2; ++d)
      acc += psum[g * N2 + d] * inv * Wfc[d * NCLS + c];
    out[i] = acc;
  }
}

extern "C" void kernel_launch(void* const* d_in, const int* in_sizes, int n_in,
                              void* d_out, int out_size, void* d_ws, size_t ws_size,
                              hipStream_t stream) {
  (void)in_sizes; (void)n_in; (void)out_size; (void)ws_size;
  const float* x    = (const float*)d_in[0];
  const int*   ei   = (const int*)d_in[1];    // [2, N_EDGES], int32 (JAX x64 off)
  const int*   batch= (const int*)d_in[2];
  const float* W1   = (const float*)d_in[3];
  const float* b1   = (const float*)d_in[4];
  const float* W2   = (const float*)d_in[5];
  const float* b2   = (const float*)d_in[6];
  const float* Wfc  = (const float*)d_in[7];
  const float* bfc  = (const float*)d_in[8];
  float* out = (float*)d_out;

  float* ws = (float*)d_ws;
  size_t off = 0;
  float* dis  = ws + off; off += N_NODES;
  float* h1   = ws + off; off += (size_t)N_NODES * N1;
  float* agg1 = ws + off; off += (size_t)N_NODES * N1;
  float* h2   = ws + off; off += (size_t)N_NODES * N2;
  float* agg2 = ws + off; off += (size_t)N_NODES * N2;
  float* psum = ws + off; off += (size_t)NGRAPH * N2;
  float* pcnt = ws + off; off += NGRAPH;

  // Re-init everything receiving atomics (deterministic across graph replays).
  k_fill4<<<256, 256, 0, stream>>>(dis, (long)N_NODES / 4, 1.0f);  // self-loop
  k_fill4<<<2048, 256, 0, stream>>>(agg1, (long)N_NODES * N1 / 4, 0.0f);
  k_fill4<<<1024, 256, 0, stream>>>(agg2, (long)N_NODES * N2 / 4, 0.0f);
  k_fill4<<<16, 256, 0, stream>>>(psum, (long)NGRAPH * N2 / 4, 0.0f);
  k_fill4<<<1, 64, 0, stream>>>(pcnt, (long)NGRAPH / 4, 0.0f);

  // Degree -> deg^-1/2 (in place)
  k_degree<<<4096, 256, 0, stream>>>(ei, dis);
  k_invsqrt<<<(N_NODES + 255) / 256, 256, 0, stream>>>(dis);

  // Layer 1: h1 = x @ W1; scatter; self-loop + bias + ReLU (into agg1)
  {
    int waves = (N_NODES / 16) * (N1 / 16);
    k_gemm_wmma<<<(waves * 32 + 255) / 256, 256, 0, stream>>>(x, W1, h1,
                                                              N_NODES, D_FEAT, N1);
  }
  k_scatter<N1><<<8192, 256, 0, stream>>>(ei, dis, h1, agg1);
  k_finalize<N1, true><<<8192, 256, 0, stream>>>(h1, dis, b1, agg1);

  // Layer 2: h2 = relu(agg1) @ W2; scatter; self-loop + bias (into agg2)
  {
    int waves = (N_NODES / 16) * (N2 / 16);
    k_gemm_wmma<<<(waves * 32 + 255) / 256, 256, 0, stream>>>(agg1, W2, h2,
                                                              N_NODES, N1, N2);
  }
  k_scatter<N2><<<8192, 256, 0, stream>>>(ei, dis, h2, agg2);
  k_finalize<N2, false><<<4096, 256, 0, stream>>>(h2, dis, b2, agg2);

  // Global mean pool + FC
  k_pool_cnt<<<(N_NODES + 255) / 256, 256, 0, stream>>>(batch, pcnt);
  {
    long n = (long)N_NODES * N2;
    k_pool_sum<<<(unsigned)((n + 255) / 256), 256, 0, stream>>>(batch, agg2, psum);
  }
  k_fc<<<(NGRAPH * NCLS + 255) / 256, 256, 0, stream>>>(psum, pcnt, Wfc, bfc, out);
}